// SmokeNoise_50087908606628
// MI455X (gfx1250) — compile-verified
//
#include <hip/hip_runtime.h>
#include <hip/hip_bf16.h>

// ---------------------------------------------------------------------------
// Vector types: WMMA f32 16x16x4 operands + TDM descriptor groups.
// ---------------------------------------------------------------------------
typedef __attribute__((ext_vector_type(2))) float v2f;
typedef __attribute__((ext_vector_type(8))) float v8f;
typedef __attribute__((ext_vector_type(4))) unsigned int u32x4;
typedef __attribute__((ext_vector_type(8))) int i32x8;
typedef __attribute__((ext_vector_type(4))) int i32x4;

// ---------------------------------------------------------------------------
// Threefry2x32-20 (exact JAX port)
// ---------------------------------------------------------------------------
__device__ __forceinline__ unsigned tf_rotl(unsigned x, int r) {
  return (x << r) | (x >> (32 - r));
}

__device__ void threefry2x32(unsigned k0, unsigned k1, unsigned x0, unsigned x1,
                             unsigned* o0, unsigned* o1) {
  const unsigned ks2 = k0 ^ k1 ^ 0x1BD11BDAu;
  const int ra[4] = {13, 15, 26, 6};
  const int rb[4] = {17, 29, 16, 24};
  x0 += k0; x1 += k1;
#pragma unroll
  for (int i = 0; i < 4; ++i) { x0 += x1; x1 = tf_rotl(x1, ra[i]); x1 ^= x0; }
  x0 += k1; x1 += ks2 + 1u;
#pragma unroll
  for (int i = 0; i < 4; ++i) { x0 += x1; x1 = tf_rotl(x1, rb[i]); x1 ^= x0; }
  x0 += ks2; x1 += k0 + 2u;
#pragma unroll
  for (int i = 0; i < 4; ++i) { x0 += x1; x1 = tf_rotl(x1, ra[i]); x1 ^= x0; }
  x0 += k0; x1 += k1 + 3u;
#pragma unroll
  for (int i = 0; i < 4; ++i) { x0 += x1; x1 = tf_rotl(x1, rb[i]); x1 ^= x0; }
  x0 += k1; x1 += ks2 + 4u;
#pragma unroll
  for (int i = 0; i < 4; ++i) { x0 += x1; x1 = tf_rotl(x1, ra[i]); x1 ^= x0; }
  x0 += ks2; x1 += k0 + 5u;
  *o0 = x0; *o1 = x1;
}

// bits[j] of jax threefry random bits over iota counts of length n
// (odd-length zero padding, split-in-half block layout).
__device__ unsigned tf_randbits(unsigned k0, unsigned k1, unsigned n, unsigned j) {
  unsigned m = (n + 1u) >> 1;
  unsigned blk, word;
  if (j < m) { blk = j; word = 0u; } else { blk = j - m; word = 1u; }
  unsigned x0 = blk;
  unsigned x1 = (m + blk < n) ? (m + blk) : 0u;
  unsigned o0, o1;
  threefry2x32(k0, k1, x0, x1, &o0, &o1);
  return word ? o1 : o0;
}

__device__ __forceinline__ void tf_fold(unsigned k0, unsigned k1, unsigned data,
                                        unsigned* o0, unsigned* o1) {
  threefry2x32(k0, k1, 0u, data, o0, o1);
}

__device__ void tf_subkey(unsigned k0, unsigned k1, unsigned nkeys, unsigned j,
                          unsigned* s0, unsigned* s1) {
  *s0 = tf_randbits(k0, k1, 2u * nkeys, 2u * j);
  *s1 = tf_randbits(k0, k1, 2u * nkeys, 2u * j + 1u);
}

__device__ float tf_uniform(unsigned k0, unsigned k1, unsigned n, unsigned j,
                            float lo, float hi) {
  unsigned bits = tf_randbits(k0, k1, n, j);
  float u = __uint_as_float((bits >> 9) | 0x3f800000u) - 1.0f;
  float v = u * (hi - lo) + lo;
  return fmaxf(lo, v);
}

// ---------------------------------------------------------------------------
// Workspace layout (floats). HM region is reused as GEMM-1 output T once HMt
// (transpose) has been produced.
// ---------------------------------------------------------------------------
#define HM_LD   528
#define HM_N    513
#define OUT_N   512
#define KPAD    528
#define OFF_HM    0                         // 528*528 : HM, later T (512x528)
#define OFF_HMT   (OFF_HM + 528 * 528)      // 528*528 : HM transposed
#define OFF_W     (OFF_HMT + 528 * 528)     // 512*528 : resize weights
#define OFF_RN    (OFF_W + 512 * 528)       // 512*512 : resized raw noise
#define OFF_CENT  (OFF_RN + 512 * 512)      // 256*256 : per-level centers
#define OFF_STATS (OFF_CENT + 256 * 256)    // [min, max]

// ---------------------------------------------------------------------------
// 1) Init heightmap: zero padded 528x528, seed corners
//    (key = fold_in(key(42), 1000); uniform((4,), -1, 1)).
// ---------------------------------------------------------------------------
__global__ void ds_init_kernel(float* __restrict__ HM) {
  int idx = blockIdx.x * blockDim.x + threadIdx.x;
  if (idx >= 528 * 528) return;
  int r = idx / 528, c = idx % 528;
  float v = 0.0f;
  int corner = -1;
  if (r == 0 && c == 0) corner = 0;
  else if (r == 0 && c == HM_N - 1) corner = 1;
  else if (r == HM_N - 1 && c == 0) corner = 2;
  else if (r == HM_N - 1 && c == HM_N - 1) corner = 3;
  if (corner >= 0) {
    unsigned ck0, ck1;
    tf_fold(0u, 42u, 1000u, &ck0, &ck1);
    v = tf_uniform(ck0, ck1, 4u, (unsigned)corner, -1.0f, 1.0f);
  }
  HM[r * HM_LD + c] = v;
}

// ---------------------------------------------------------------------------
// 2) Diamond-square per level: centers pass then edges pass.
// ---------------------------------------------------------------------------
__global__ void ds_centers_kernel(float* __restrict__ HM, float* __restrict__ CENT,
                                  int level) {
  int g = 1 << level;
  int idx = blockIdx.x * blockDim.x + threadIdx.x;
  if (idx >= g * g) return;
  int i = idx / g, j = idx % g;
  int L = 512 >> level, h = L >> 1;
  int li = i * L, hi2 = li + L, mi = li + h;
  int lj = j * L, hj = lj + L, mj = lj + h;
  float nw = HM[li * HM_LD + lj];
  float ne = HM[li * HM_LD + hj];
  float sw = HM[hi2 * HM_LD + lj];
  float se = HM[hi2 * HM_LD + hj];
  unsigned lk0, lk1, s0, s1;
  tf_fold(0u, 42u, (unsigned)level, &lk0, &lk1);
  tf_subkey(lk0, lk1, 5u, 0u, &s0, &s1);
  float r = 1.0f / (float)(level + 1);
  float c = (nw + ne + sw + se) * 0.25f +
            tf_uniform(s0, s1, (unsigned)(g * g), (unsigned)idx, -r, r);
  HM[mi * HM_LD + mj] = c;
  CENT[i * 256 + j] = c;
}

__global__ void ds_edges_kernel(float* __restrict__ HM, const float* __restrict__ CENT,
                                int level) {
  int g = 1 << level;
  int idx = blockIdx.x * blockDim.x + threadIdx.x;
  if (idx >= g * g) return;
  int i = idx / g, j = idx % g;
  int L = 512 >> level, h = L >> 1;
  int li = i * L, hi2 = li + L, mi = li + h;
  int lj = j * L, hj = lj + L, mj = lj + h;
  float c = CENT[i * 256 + j];
  float ne = HM[li * HM_LD + hj];
  float sw = HM[hi2 * HM_LD + lj];
  float se = HM[hi2 * HM_LD + hj];
  unsigned lk0, lk1;
  tf_fold(0u, 42u, (unsigned)level, &lk0, &lk1);
  float r = 1.0f / (float)(level + 1);
  unsigned n2 = (unsigned)(g * g);

  unsigned k3a, k3b, k4a, k4b;
  tf_subkey(lk0, lk1, 5u, 3u, &k3a, &k3b);
  tf_subkey(lk0, lk1, 5u, 4u, &k4a, &k4b);

  // interior temp resolves to own center (reference's double-count)
  float tr = (j < g - 1) ? c : CENT[i * 256 + 0];
  float right = (c + ne + se + tr) * 0.25f +
                tf_uniform(k3a, k3b, n2, (unsigned)idx, -r, r);
  HM[mi * HM_LD + hj] = right;

  float tb = (i < g - 1) ? c : CENT[0 * 256 + j];
  float bottom = (c + sw + se + tb) * 0.25f +
                 tf_uniform(k4a, k4b, n2, (unsigned)idx, -r, r);
  HM[hi2 * HM_LD + mj] = bottom;

  if (i == 0) {
    unsigned k1a, k1b;
    tf_subkey(lk0, lk1, 5u, 1u, &k1a, &k1b);
    float top = (CENT[j] + HM[lj] + HM[hj] + CENT[(g - 1) * 256 + j]) * 0.25f +
                tf_uniform(k1a, k1b, (unsigned)g, (unsigned)j, -r, r);
    HM[mj] = top;
  }
  if (j == 0) {
    unsigned k2a, k2b;
    tf_subkey(lk0, lk1, 5u, 2u, &k2a, &k2b);
    float left = (CENT[i * 256] + HM[li * HM_LD] + HM[hi2 * HM_LD] +
                  CENT[i * 256 + (g - 1)]) * 0.25f +
                 tf_uniform(k2a, k2b, (unsigned)g, (unsigned)i, -r, r);
    HM[mi * HM_LD] = left;
  }
}

// ---------------------------------------------------------------------------
// 3) min/max over the 513x513 heightmap (single-block LDS tree reduce).
// ---------------------------------------------------------------------------
__global__ __launch_bounds__(1024) void minmax_kernel(const float* __restrict__ HM,
                                                      float* __restrict__ stats) {
  __shared__ float smin[1024];
  __shared__ float smax[1024];
  float mn = 3.4e38f, mx = -3.4e38f;
  for (int idx = threadIdx.x; idx < HM_N * HM_N; idx += 1024) {
    float v = HM[(idx / HM_N) * HM_LD + (idx % HM_N)];
    mn = fminf(mn, v);
    mx = fmaxf(mx, v);
  }
  smin[threadIdx.x] = mn;
  smax[threadIdx.x] = mx;
  __syncthreads();
  for (int s = 512; s > 0; s >>= 1) {
    if ((int)threadIdx.x < s) {
      smin[threadIdx.x] = fminf(smin[threadIdx.x], smin[threadIdx.x + s]);
      smax[threadIdx.x] = fmaxf(smax[threadIdx.x], smax[threadIdx.x + s]);
    }
    __syncthreads();
  }
  if (threadIdx.x == 0) { stats[0] = smin[0]; stats[1] = smax[0]; }
}

// ---------------------------------------------------------------------------
// 4) Antialiased bilinear resize weight matrix W (512 x 513, K-padded to 528).
// ---------------------------------------------------------------------------
__global__ void gen_w_kernel(float* __restrict__ W) {
  int i = blockIdx.x * blockDim.x + threadIdx.x;
  if (i >= OUT_N) return;
  const float inv_scale = 513.0f / 512.0f;
  float x = ((float)i + 0.5f) * inv_scale - 0.5f;
  float wsum = 0.0f;
  for (int k = 0; k < HM_N; ++k)
    wsum += fmaxf(0.0f, 1.0f - fabsf((float)k - x) / inv_scale);
  float inv = 1.0f / wsum;
  for (int k = 0; k < KPAD; ++k) {
    float w = (k < HM_N)
                  ? fmaxf(0.0f, 1.0f - fabsf((float)k - x) / inv_scale) * inv
                  : 0.0f;
    W[i * KPAD + k] = w;
  }
}

// ---------------------------------------------------------------------------
// 5) Transpose HM -> HMt so both GEMMs read B along K contiguously (b64).
// ---------------------------------------------------------------------------
__global__ void transpose_kernel(const float* __restrict__ src,
                                 float* __restrict__ dst) {
  int idx = blockIdx.x * blockDim.x + threadIdx.x;
  if (idx >= 528 * 528) return;
  int r = idx / 528, c = idx % 528;
  dst[c * 528 + r] = src[r * 528 + c];
}

// ---------------------------------------------------------------------------
// 6) WMMA f32 GEMM, B pre-transposed: C(tm:tm+16, tn:tn+16) over K.
//    A-layout (ISA 7.12.2): lanes 0-15 hold K 0/1 in v[0]/v[1], lanes 16-31
//    hold K 2/3.  D: VGPR r -> M=r (lanes 0-15) / M=r+8 (lanes 16-31).
//    Branch-free inner loop: 2x global_load_b64 + v_wmma per K-step.
// ---------------------------------------------------------------------------
__global__ __launch_bounds__(32) void wmma_gemm_bt_kernel(
    const float* __restrict__ A, const float* __restrict__ Bt,
    float* __restrict__ C, int ntiles_n, int K, int lda, int ldb, int ldc) {
  int tile = blockIdx.x;
  int tm = (tile / ntiles_n) * 16;
  int tn = (tile % ntiles_n) * 16;
  int lane = threadIdx.x;
  int half = lane >> 4;
  int l16 = lane & 15;

  const float* arow = A + (tm + l16) * lda + 2 * half;
  const float* bcol = Bt + (tn + l16) * ldb + 2 * half;
  v8f acc = {};
  for (int kb = 0; kb < K; kb += 4) {
    v2f a, b;
    a.x = arow[kb];
    a.y = arow[kb + 1];
    b.x = bcol[kb];
    b.y = bcol[kb + 1];
    acc = __builtin_amdgcn_wmma_f32_16x16x4_f32(
        false, a, false, b, (short)0, acc, false, false);
  }
#pragma unroll
  for (int v = 0; v < 8; ++v) {
    int m = v + 8 * half;
    C[(tm + m) * ldc + tn + l16] = acc[v];
  }
}

// ---------------------------------------------------------------------------
// 7) Streaming blend. Each 256-thread block covers exactly 1024 elements =
//    one contiguous 4KB slice of the (L2-resident) noise map, shared by all
//    8 waves. Wave 0 stages that slice into LDS with ONE Tensor-Data-Mover
//    transfer (D# built in scalar regs), waits on TENSORcnt, and the block
//    barrier publishes it — freeing the per-lane VMEM path for the img
//    stream. Grid*1024 == total elements, so no thread exits early.
// ---------------------------------------------------------------------------
__global__ __launch_bounds__(256) void blend_kernel(
    const float* __restrict__ img, const float* __restrict__ rn_raw,
    const float* __restrict__ stats, float* __restrict__ out) {
  __shared__ float lds_rn[1024];

  if (threadIdx.x < 32) {  // wave 0 only (wave32)
    unsigned rn_off = (blockIdx.x & 255u) << 10;  // in-plane float offset
    unsigned long long ga =
        (unsigned long long)(uintptr_t)(rn_raw + rn_off);
    u32x4 g0;
    g0[0] = 1u;                                    // count=1 (user mode)
    g0[1] = 0u;                                    // lds_addr = 0 bytes
    g0[2] = (unsigned)(ga & 0xffffffffull);        // global_addr[31:0]
    g0[3] = (unsigned)((ga >> 32) & 0x1ffffffull)  // global_addr[56:32]
            | (2u << 30);                          // type = 2 ("image")
    i32x8 g1 = {0, 0, 0, 0, 0, 0, 0, 0};
    g1[0] = 2 << 16;                   // data_size = 4 bytes
    g1[1] = 0;                         // tensor_dim0[15:0] (0x40000) << 16 = 0
    g1[2] = (int)(262144u >> 16)       // tensor_dim0[31:16]
            | (1 << 16);               // tensor_dim1 = 1
    g1[3] = (int)(1024u << 16);        // tile_dim0 = 1024 elements
    g1[4] = 1;                         // tile_dim1 = 1
    g1[5] = 262144;                    // tensor_dim0_stride
    i32x4 gz = {0, 0, 0, 0};
#if __has_include(<hip/amd_detail/amd_gfx1250_TDM.h>)
    i32x8 gz8 = {0, 0, 0, 0, 0, 0, 0, 0};
    __builtin_amdgcn_tensor_load_to_lds(g0, g1, gz, gz, gz8, 0);
#else
    __builtin_amdgcn_tensor_load_to_lds(g0, g1, gz, gz, 0);
#endif
    __builtin_amdgcn_s_wait_tensorcnt(0);
  }
  __syncthreads();

  int idx = blockIdx.x * 256 + threadIdx.x;  // float4 index
  const float4* im4 = (const float4*)img;
  float4* o4 = (float4*)out;

  __builtin_prefetch(&im4[idx + 4096], 0, 0);

  float mn = stats[0];
  float sc = 0.5f / (stats[1] - mn);
  float4 nr = ((const float4*)lds_rn)[threadIdx.x];
  float4 iv = im4[idx];
  float r0 = (nr.x - mn) * sc, r1 = (nr.y - mn) * sc;
  float r2 = (nr.z - mn) * sc, r3 = (nr.w - mn) * sc;
  float4 ov;
  ov.x = fminf(1.0f, fmaxf(0.0f, iv.x * (1.0f - r0) + r0));
  ov.y = fminf(1.0f, fmaxf(0.0f, iv.y * (1.0f - r1) + r1));
  ov.z = fminf(1.0f, fmaxf(0.0f, iv.z * (1.0f - r2) + r2));
  ov.w = fminf(1.0f, fmaxf(0.0f, iv.w * (1.0f - r3) + r3));
  o4[idx] = ov;
}

// ---------------------------------------------------------------------------
// Host-side launch chain (all on `stream`, deterministic, capture-safe).
// ---------------------------------------------------------------------------
extern "C" void kernel_launch(void* const* d_in, const int* in_sizes, int n_in,
                              void* d_out, int out_size, void* d_ws, size_t ws_size,
                              hipStream_t stream) {
  (void)in_sizes; (void)n_in; (void)out_size; (void)ws_size;
  const float* img = (const float*)d_in[0];
  float* out = (float*)d_out;
  float* ws = (float*)d_ws;

  float* HM = ws + OFF_HM;        // heightmap; reused as T after transpose
  float* HMT = ws + OFF_HMT;      // heightmap transposed
  float* W = ws + OFF_W;          // resize weights
  float* RN = ws + OFF_RN;        // resized raw noise (512x512)
  float* CENT = ws + OFF_CENT;    // per-level center scratch
  float* STATS = ws + OFF_STATS;  // [min, max]
  float* T = HM;                  // GEMM-1 output aliases HM (read via HMT)

  // 1) init + corners
  ds_init_kernel<<<(528 * 528 + 255) / 256, 256, 0, stream>>>(HM);

  // 2) nine diamond-square levels
  for (int level = 0; level < 9; ++level) {
    int g = 1 << level;
    int n = g * g;
    int blk = n < 256 ? n : 256;
    int grid = (n + blk - 1) / blk;
    ds_centers_kernel<<<grid, blk, 0, stream>>>(HM, CENT, level);
    ds_edges_kernel<<<grid, blk, 0, stream>>>(HM, CENT, level);
  }

  // 3) min/max for normalization (folded into blend)
  minmax_kernel<<<1, 1024, 0, stream>>>(HM, STATS);

  // 4) interpolation weight matrix
  gen_w_kernel<<<2, 256, 0, stream>>>(W);

  // 5) transpose, then resize as two WMMA f32 GEMMs: T = W*HM ; RN = T*W^T
  transpose_kernel<<<(528 * 528 + 255) / 256, 256, 0, stream>>>(HM, HMT);
  wmma_gemm_bt_kernel<<<(512 / 16) * (KPAD / 16), 32, 0, stream>>>(
      W, HMT, T, KPAD / 16, KPAD, KPAD, KPAD, KPAD);
  wmma_gemm_bt_kernel<<<(512 / 16) * (512 / 16), 32, 0, stream>>>(
      T, W, RN, 512 / 16, KPAD, KPAD, KPAD, 512);

  // 6) bandwidth-bound blend over (32,3,512,512): 24576 blocks x 1024 elems
  int total4 = 32 * 3 * 512 * 512 / 4;
  blend_kernel<<<total4 / 256, 256, 0, stream>>>(img, RN, STATS, out);
}